// TorchSolver_12541304504732
// MI455X (gfx1250) — compile-verified
//
#include <hip/hip_runtime.h>

#define N_GENOMES 30000
#define N_GENES   240000
#define N_SAMPLES 128
#define N_SEQS    80000

// ---------------- CDNA5 async global->LDS support (compile-guarded) ----------------
#if defined(__AMDGCN__) && defined(__gfx1250__) && __has_builtin(__builtin_amdgcn_global_load_async_to_lds_b128)
#define ASYNC_LDS 1

typedef int v4i __attribute__((vector_size(16)));
typedef __attribute__((address_space(1))) v4i* g_v4i_p;  // global (prints as __device__)
typedef __attribute__((address_space(3))) v4i* l_v4i_p;  // LDS

__device__ __forceinline__ void async_copy16(const void* g, void* l) {
  // (global src, lds dst, imm offset, imm cpol)
  __builtin_amdgcn_global_load_async_to_lds_b128((g_v4i_p)g, (l_v4i_p)l, 0, 0);
}

template <int N>
__device__ __forceinline__ void wait_async() {
#if __has_builtin(__builtin_amdgcn_s_wait_asynccnt)
  __builtin_amdgcn_s_wait_asynccnt(N);
  asm volatile("" ::: "memory");  // keep LDS reads below the wait
#else
  asm volatile("s_wait_asynccnt %0" ::"i"(N) : "memory");
#endif
}
#else
#define ASYNC_LDS 0
#endif

// ---------------- Pass 5 (defined first so the disasm snippet shows the hot kernel):
// one wave32 per sequence, async-staged rows, accumulate + single b128 store ----------------
__global__ __launch_bounds__(256) void seq_accum_kernel(
    const float4* __restrict__ A4, const float4* __restrict__ B4,
    const float* __restrict__ pos, const int* __restrict__ gidx,
    const int* __restrict__ offs, const int* __restrict__ glist,
    float4* __restrict__ out4) {
  const int wave = threadIdx.x >> 5;
  const int lane = threadIdx.x & 31;
  const int k    = blockIdx.x * 8 + wave;
  if (k >= N_SEQS) return;

  const int start = offs[k];
  const int end   = offs[k + 1];
  const int n     = end - start;
  float4 acc{0.f, 0.f, 0.f, 0.f};

  // Deterministic consumption order: ascending gene id (glist order is atomic-scatter order).
  // Segments are tiny (avg 3 genes) — keep this loop rolled.
  auto next_gene = [&](int prev) -> int {
    int mn = 0x7fffffff;
#pragma clang loop unroll(disable)
    for (int j = start; j < end; ++j) {
      int g = glist[j];
      if (g > prev && g < mn) mn = g;
    }
    return mn;
  };

#if ASYNC_LDS
  // 8 waves x 2 stages x {A,B} x 128 floats = 16 KB LDS, double-buffered async pipeline.
  __shared__ __align__(16) float lds[8][2][2][N_SAMPLES];
  int cur = -1;
  if (n > 0) {
    cur = next_gene(-1);
    int row = gidx[cur];
    async_copy16(&A4[row * 32 + lane], &lds[wave][0][0][lane * 4]);
    async_copy16(&B4[row * 32 + lane], &lds[wave][0][1][lane * 4]);
  }
#pragma clang loop unroll(disable)
  for (int i = 0; i < n; ++i) {
    const int g = cur;
    if (i + 1 < n) {
      cur = next_gene(g);
      int row = gidx[cur];
      const int st = (i + 1) & 1;
      async_copy16(&A4[row * 32 + lane], &lds[wave][st][0][lane * 4]);
      async_copy16(&B4[row * 32 + lane], &lds[wave][st][1][lane * 4]);
      wait_async<2>();  // current stage's 2 transfers complete; next stage may be in flight
    } else {
      wait_async<0>();
    }
    const float  p = pos[g];
    const float4 a = *(const float4*)&lds[wave][i & 1][0][lane * 4];
    const float4 b = *(const float4*)&lds[wave][i & 1][1][lane * 4];
    acc.x += __expf(a.x + 1.f - p * b.x);
    acc.y += __expf(a.y + 1.f - p * b.y);
    acc.z += __expf(a.z + 1.f - p * b.z);
    acc.w += __expf(a.w + 1.f - p * b.w);
  }
#else
  int last = -1;
#pragma clang loop unroll(disable)
  for (int i = 0; i < n; ++i) {
    const int g = next_gene(last);
    last = g;
    const int   row = gidx[g];
    const float p   = pos[g];
    const float4 a  = A4[row * 32 + lane];  // global_load_b128, L2-resident
    const float4 b  = B4[row * 32 + lane];
    acc.x += __expf(a.x + 1.f - p * b.x);
    acc.y += __expf(a.y + 1.f - p * b.y);
    acc.z += __expf(a.z + 1.f - p * b.z);
    acc.w += __expf(a.w + 1.f - p * b.w);
  }
#endif

  out4[k * 32 + lane] = acc;  // single coalesced b128 store; no float atomics anywhere
}

// ---------------- Pass 1: zero per-seq counters ----------------
__global__ __launch_bounds__(256) void zero_kernel(int* __restrict__ cnt) {
  int i = blockIdx.x * blockDim.x + threadIdx.x;
  if (i < N_SEQS) cnt[i] = 0;
}

// ---------------- Pass 2: histogram of genes per sequence ----------------
__global__ __launch_bounds__(256) void hist_kernel(const int* __restrict__ seq,
                                                   int* __restrict__ cnt) {
  int g = blockIdx.x * blockDim.x + threadIdx.x;
  if (g < N_GENES) atomicAdd(&cnt[seq[g]], 1);
}

// ---------------- Pass 3: single-block exclusive scan -> offsets + cursors ----------------
// cnt and cursor may alias: each slot is read (count) before it is overwritten (cursor).
__global__ __launch_bounds__(1024) void scan_kernel(const int* cnt,
                                                    int* __restrict__ offs,
                                                    int* cursor) {
  __shared__ int s[1024];
  const int t  = threadIdx.x;
  const int CH = (N_SEQS + 1023) / 1024;  // 79
  const int lo = t * CH;
  const int hi = (lo + CH < N_SEQS) ? lo + CH : N_SEQS;

  int sum = 0;
  for (int i = lo; i < hi; ++i) sum += cnt[i];
  s[t] = sum;
  __syncthreads();
  // Hillis-Steele inclusive scan over 1024 partials
  for (int off = 1; off < 1024; off <<= 1) {
    int v = (t >= off) ? s[t - off] : 0;
    __syncthreads();
    s[t] += v;
    __syncthreads();
  }
  int base = s[t] - sum;  // exclusive prefix for this chunk
  for (int i = lo; i < hi; ++i) {
    int c = cnt[i];
    offs[i]   = base;
    cursor[i] = base;  // may overwrite cnt[i]; c already read
    base += c;
  }
  if (t == 0) offs[N_SEQS] = N_GENES;
}

// ---------------- Pass 4: scatter gene ids into CSR list ----------------
__global__ __launch_bounds__(256) void scatter_kernel(const int* __restrict__ seq,
                                                      int* __restrict__ cursor,
                                                      int* __restrict__ glist) {
  int g = blockIdx.x * blockDim.x + threadIdx.x;
  if (g < N_GENES) {
    int p = atomicAdd(&cursor[seq[g]], 1);
    glist[p] = g;
  }
}

// ---------------- Host-side launch ----------------
extern "C" void kernel_launch(void* const* d_in, const int* in_sizes, int n_in,
                              void* d_out, int out_size, void* d_ws, size_t ws_size,
                              hipStream_t stream) {
  const float* A    = (const float*)d_in[0];   // (30000,128)
  const float* B    = (const float*)d_in[1];   // (30000,128)
  const float* pos  = (const float*)d_in[2];   // (240000,)
  const int*   gidx = (const int*)d_in[3];     // (240000,)
  const int*   sidx = (const int*)d_in[4];     // (240000,)

  int* cnt   = (int*)d_ws;           // N_SEQS   (counts, then reused as scatter cursors)
  int* offs  = cnt + N_SEQS;         // N_SEQS+1 (CSR offsets)
  int* glist = offs + (N_SEQS + 1);  // N_GENES  (CSR gene list)

  zero_kernel<<<(N_SEQS + 255) / 256, 256, 0, stream>>>(cnt);
  hist_kernel<<<(N_GENES + 255) / 256, 256, 0, stream>>>(sidx, cnt);
  scan_kernel<<<1, 1024, 0, stream>>>(cnt, offs, cnt /*cursor, aliases cnt*/);
  scatter_kernel<<<(N_GENES + 255) / 256, 256, 0, stream>>>(sidx, cnt, glist);
  seq_accum_kernel<<<N_SEQS / 8, 256, 0, stream>>>(
      (const float4*)A, (const float4*)B, pos, gidx, offs, glist, (float4*)d_out);
}